// NN_4260607557798
// MI455X (gfx1250) — compile-verified
//
#include <hip/hip_runtime.h>
#include <math.h>

// ---------------- problem constants (from reference) ----------------
#define B_   16
#define N_   2048
#define K_   20
#define M_   (B_ * N_)        // 32768 points total
#define C1_  64               // edge-conv output channels
#define EPS_ 1e-5f

typedef __attribute__((ext_vector_type(2))) float v2f;
typedef __attribute__((ext_vector_type(8))) float v8f;

// =====================================================================
// 0) zero the BN statistics accumulators (harness does not re-zero ws)
// =====================================================================
__global__ void zero_stats_kernel(float* stats) {
    int t = threadIdx.x;
    if (t < 128) stats[t] = 0.0f;   // [0:64)=sum, [64:128)=sumsq
}

// =====================================================================
// 1) KNN: for each point, top-20 smallest squared distances (incl self)
//    One block = 256 query points of one batch; all 2048 points of the
//    batch staged in LDS (24KB). Top-20 kept sorted in registers.
// =====================================================================
__global__ void knn_kernel(const float* __restrict__ x, int* __restrict__ idx) {
    __shared__ float spts[N_ * 3];
    const int b   = blockIdx.y;
    const int tid = threadIdx.x;

    // cooperative stage: batch points (x[...,0:3]) -> LDS
    for (int j = tid; j < N_; j += 256) {
        const float* p = x + ((size_t)(b * N_ + j)) * 6;
        __builtin_prefetch(p + 256 * 6, 0, 0);   // global_prefetch_b8
        spts[j * 3 + 0] = p[0];
        spts[j * 3 + 1] = p[1];
        spts[j * 3 + 2] = p[2];
    }
    __syncthreads();

    const int q  = blockIdx.x * 256 + tid;
    const float qx = spts[q * 3 + 0];
    const float qy = spts[q * 3 + 1];
    const float qz = spts[q * 3 + 2];

    float bd[K_];
    int   bi[K_];
#pragma unroll
    for (int t = 0; t < K_; ++t) { bd[t] = 3.4e38f; bi[t] = 0; }

    for (int j = 0; j < N_; ++j) {
        const float dx = spts[j * 3 + 0] - qx;
        const float dy = spts[j * 3 + 1] - qy;
        const float dz = spts[j * 3 + 2] - qz;
        const float d2 = dx * dx + dy * dy + dz * dz;
        if (d2 < bd[K_ - 1]) {        // rare path: sorted insert
            float dc = d2; int ic = j;
#pragma unroll
            for (int t = 0; t < K_; ++t) {
                if (dc < bd[t]) {
                    float td = bd[t]; bd[t] = dc; dc = td;
                    int   ti = bi[t]; bi[t] = ic; ic = ti;
                }
            }
        }
    }

    int* o = idx + ((size_t)(b * N_ + q)) * K_;
#pragma unroll
    for (int t = 0; t < K_; ++t) o[t] = bi[t];
}

// ---------------------------------------------------------------------
// shared device helper: gather the 20x6 edge-feature block for `item`
// into sfeat (called by all 64 threads of a group). Branchless: the
// self-position subtraction is an unconditional load times a mask so no
// exec-divergent load path is generated.
// ---------------------------------------------------------------------
__device__ __forceinline__ void gather_feat(const float* __restrict__ x,
                                            const int* __restrict__ idx,
                                            int item, int lane64, float* sfeat) {
    const int bbase = (item >> 11) << 11;        // b * N_
    for (int e = lane64; e < K_ * 6; e += 64) {
        const int k = e / 6, c = e % 6;
        const int jn = idx[(size_t)item * K_ + k];
        const float vn   = x[((size_t)(bbase + jn)) * 6 + c];
        const float vself = x[(size_t)item * 6 + c];
        const float mask = (c < 3 && k != 0) ? 1.0f : 0.0f;
        sfeat[k * 6 + c] = vn - mask * vself;    // edge vector (k==0: raw pos)
    }
}

// =====================================================================
// 2) edge-conv + BN statistics (sum, sumsq per channel over B*N*K)
//    512 blocks * 4 groups/block; each group of 64 threads handles 16
//    points; thread = one of the 64 output channels.
// =====================================================================
__global__ void conv_stats_kernel(const float* __restrict__ x,
                                  const int* __restrict__ idx,
                                  const float* __restrict__ w_c1,
                                  float* __restrict__ stats) {
    __shared__ float sw[C1_ * 6];
    __shared__ float sfeat[4][K_ * 6];
    __shared__ float rs[C1_], rss[C1_];

    const int tid = threadIdx.x;
    const int g   = tid >> 6;       // group 0..3
    const int o   = tid & 63;       // channel

    for (int e = tid; e < C1_ * 6; e += 256) sw[e] = w_c1[e];
    if (tid < C1_) { rs[tid] = 0.0f; rss[tid] = 0.0f; }
    __syncthreads();

    float s = 0.0f, ss = 0.0f;
    const int group_global = blockIdx.x * 4 + g;    // 0..2047
    for (int it = 0; it < 16; ++it) {
        const int item = group_global * 16 + it;    // 0..32767
        gather_feat(x, idx, item, o, sfeat[g]);
        __syncthreads();
#pragma unroll 4
        for (int k = 0; k < K_; ++k) {
            float h = 0.0f;
#pragma unroll
            for (int c = 0; c < 6; ++c) h += sw[o * 6 + c] * sfeat[g][k * 6 + c];
            s  += h;
            ss += h * h;
        }
        __syncthreads();
    }
    atomicAdd(&rs[o],  s);
    atomicAdd(&rss[o], ss);
    __syncthreads();
    if (tid < C1_) {
        atomicAdd(&stats[tid],        rs[tid]);
        atomicAdd(&stats[C1_ + tid],  rss[tid]);
    }
}

// =====================================================================
// 3) fold BN into per-channel scale/shift: stats[128+o], stats[192+o]
// =====================================================================
__global__ void bn_finalize_kernel(float* stats, const float* __restrict__ g1,
                                   const float* __restrict__ be1) {
    const int o = threadIdx.x;
    if (o < C1_) {
        const float cnt = (float)((size_t)M_ * K_);
        const float mu  = stats[o] / cnt;
        const float var = stats[C1_ + o] / cnt - mu * mu;
        const float sc  = g1[o] * rsqrtf(var + EPS_);
        stats[128 + o] = sc;
        stats[192 + o] = be1[o] - mu * sc;
    }
}

// =====================================================================
// 4) recompute edge-conv, apply BN affine + leaky-ReLU, max over K
//    8192 blocks * 4 groups; one point per group; thread = channel.
// =====================================================================
__global__ void edge_pool_kernel(const float* __restrict__ x,
                                 const int* __restrict__ idx,
                                 const float* __restrict__ w_c1,
                                 const float* __restrict__ stats,
                                 float* __restrict__ h0) {
    __shared__ float sw[C1_ * 6];
    __shared__ float sfeat[4][K_ * 6];

    const int tid = threadIdx.x;
    const int g   = tid >> 6;
    const int o   = tid & 63;

    for (int e = tid; e < C1_ * 6; e += 256) sw[e] = w_c1[e];
    __syncthreads();

    const int item = blockIdx.x * 4 + g;
    gather_feat(x, idx, item, o, sfeat[g]);
    __syncthreads();

    const float sc = stats[128 + o];
    const float sh = stats[192 + o];
    float mx = -3.4e38f;
#pragma unroll 4
    for (int k = 0; k < K_; ++k) {
        float h = 0.0f;
#pragma unroll
        for (int c = 0; c < 6; ++c) h += sw[o * 6 + c] * sfeat[g][k * 6 + c];
        h = h * sc + sh;
        h = (h > 0.0f) ? h : 0.2f * h;   // leaky relu
        mx = fmaxf(mx, h);
    }
    h0[(size_t)item * C1_ + o] = mx;
}

// =====================================================================
// 5) fp32 GEMM via V_WMMA_F32_16X16X4_F32:
//    out[M,Cout] = relu?( in[M,Cin] @ w[Cout,Cin]^T + bias )
//    Block = 8 waves sharing ONE N-tile (blockIdx.y): the zero-padded
//    16xCin weight tile + bias are staged in LDS once with contiguous
//    coalesced loads, so the hot k-loop has NO conditionals:
//      A frag: unconditional global_load_b64 per lane
//      B frag: ds_load_b64 from LDS
//    A frag (16x4, ISA 7.12.2): lane l -> row l%16, K pair 2*(l/16).
//    B frag (4x16): lane l -> col l%16 (= W row), same K pair.
//    C/D: vgpr v at lane 16h+n -> D[v+8h][n].
// =====================================================================
__global__ void gemm_wmma_kernel(const float* __restrict__ in,
                                 const float* __restrict__ w,
                                 const float* __restrict__ bias,
                                 float* __restrict__ out,
                                 int Cin, int Cout, int relu) {
    extern __shared__ float smem[];            // [16*Cin] weights + [16] bias
    float* swt   = smem;
    float* sbias = smem + 16 * Cin;

    const int tid  = threadIdx.x;
    const int lane = tid & 31;
    const int wave = tid >> 5;
    const int n0   = blockIdx.y * 16;

    // stage zero-padded weight tile: swt[r*Cin + k] = W[n0+r][k] (or 0)
    for (int e = tid; e < 16 * Cin; e += 256) {
        const int r = e / Cin, kk = e - r * Cin;
        const int gr = n0 + r;
        swt[e] = (gr < Cout) ? w[(size_t)gr * Cin + kk] : 0.0f;
    }
    if (tid < 16) {
        const int gr = n0 + tid;
        sbias[tid] = (gr < Cout) ? bias[gr] : 0.0f;
    }
    __syncthreads();

    const int m0   = (blockIdx.x * 8 + wave) * 16;   // this wave's M tile
    const int half = lane >> 4;          // 0/1 -> K pair {0,1} or {2,3}
    const int sub  = lane & 15;
    const float* arow_p = in + (size_t)(m0 + sub) * Cin;
    const float* brow_p = swt + sub * Cin;

    v8f acc = {};
#pragma unroll 4
    for (int k = 0; k < Cin; k += 4) {
        const int ka = k + half * 2;
        v2f a, bfr;
        a.x   = arow_p[ka];
        a.y   = arow_p[ka + 1];
        bfr.x = brow_p[ka];
        bfr.y = brow_p[ka + 1];
        acc = __builtin_amdgcn_wmma_f32_16x16x4_f32(
                  /*neg_a=*/false, a, /*neg_b=*/false, bfr,
                  /*c_mod=*/(short)0, acc, /*reuse_a=*/false, /*reuse_b=*/false);
    }

    const int   ncol   = n0 + sub;
    const bool  nvalid = (ncol < Cout);  // only diverges for the 13-wide head
    const float bn     = sbias[sub];
#pragma unroll
    for (int v = 0; v < 8; ++v) {
        const int m = m0 + v + half * 8;
        float val = acc[v] + bn;
        if (relu) val = fmaxf(val, 0.0f);
        if (nvalid) out[(size_t)m * Cout + ncol] = val;
    }
}

// =====================================================================
// host-side launcher
// =====================================================================
extern "C" void kernel_launch(void* const* d_in, const int* in_sizes, int n_in,
                              void* d_out, int out_size, void* d_ws, size_t ws_size,
                              hipStream_t stream) {
    const float* x    = (const float*)d_in[0];
    const float* w_c1 = (const float*)d_in[1];
    const float* g1   = (const float*)d_in[2];
    const float* be1  = (const float*)d_in[3];
    const float* w_m[6] = {(const float*)d_in[4],  (const float*)d_in[6],
                           (const float*)d_in[8],  (const float*)d_in[10],
                           (const float*)d_in[12], (const float*)d_in[14]};
    const float* b_m[6] = {(const float*)d_in[5],  (const float*)d_in[7],
                           (const float*)d_in[9],  (const float*)d_in[11],
                           (const float*)d_in[13], (const float*)d_in[15]};
    float* out = (float*)d_out;

    // workspace carve-up
    char*  ws  = (char*)d_ws;
    size_t off = 0;
    int*   idx   = (int*)(ws + off);   off += (size_t)M_ * K_ * sizeof(int);
    float* stats = (float*)(ws + off); off += 256 * sizeof(float);
    float* h0    = (float*)(ws + off); off += (size_t)M_ * C1_ * sizeof(float);
    float* bufA  = (float*)(ws + off); off += (size_t)M_ * 256 * sizeof(float);
    float* bufB  = (float*)(ws + off); off += (size_t)M_ * 256 * sizeof(float);
    (void)ws_size; (void)in_sizes; (void)n_in; (void)out_size;

    zero_stats_kernel<<<1, 128, 0, stream>>>(stats);
    knn_kernel<<<dim3(N_ / 256, B_), 256, 0, stream>>>(x, idx);
    conv_stats_kernel<<<512, 256, 0, stream>>>(x, idx, w_c1, stats);
    bn_finalize_kernel<<<1, 64, 0, stream>>>(stats, g1, be1);
    edge_pool_kernel<<<M_ / 4, 256, 0, stream>>>(x, idx, w_c1, stats, h0);

    // MLP chain: h0(64) ->A64 ->B128 ->A256 ->B128 ->A64 -> out(13)
    struct L { const float* in; int li; float* o; int cin; int cout; int relu; };
    const L layers[6] = {
        { h0,   0, bufA, 64,  64,  1 },
        { bufA, 1, bufB, 64,  128, 1 },
        { bufB, 2, bufA, 128, 256, 1 },
        { bufA, 3, bufB, 256, 128, 1 },
        { bufB, 4, bufA, 128, 64,  1 },
        { bufA, 5, out,  64,  13,  0 },
    };
    for (int i = 0; i < 6; ++i) {
        const int tiles_m = M_ / 16;                       // 2048, div by 8
        const int tiles_n = (layers[i].cout + 15) / 16;
        const size_t shmem = (16 * layers[i].cin + 16) * sizeof(float);
        gemm_wmma_kernel<<<dim3(tiles_m / 8, tiles_n), 256, shmem, stream>>>(
            layers[i].in, w_m[layers[i].li], b_m[layers[i].li], layers[i].o,
            layers[i].cin, layers[i].cout, layers[i].relu);
    }
}